// Mix3DEmbeddingV5_6949257085361
// MI455X (gfx1250) — compile-verified
//
#include <hip/hip_runtime.h>
#include <math.h>

// MI455X / gfx1250, wave32. fp32 GEMMs via V_WMMA_F32_16X16X4_F32.
typedef __attribute__((ext_vector_type(2))) float v2f;
typedef __attribute__((ext_vector_type(8))) float v8f;

#define BS 4
#define L 2048
#define D 512
#define ED 1024
#define TROWS 1024           // time values are ints in [0,1010)
#define AST 516              // padded LDS stride for 512-wide tiles (516%64=4 -> conflict-free col reads)
#define HST 132              // padded LDS stride for 128-wide tile

__device__ __forceinline__ float gelu_exact(float x) {
  return 0.5f * x * (1.0f + erff(x * 0.70710678118654752440f));
}

__device__ __forceinline__ v8f wmma4(v2f a, v2f b, v8f c) {
  // D = A(16x4,f32) * B(4x16,f32) + C(16x16,f32)
  return __builtin_amdgcn_wmma_f32_16x16x4_f32(false, a, false, b, (short)0, c, false, false);
}

// ---------------------------------------------------------------------------
// K0: temb_table[t][d] = MLP(sinusoidal(t)) for t = 0..1023.
// One block = 16 t-rows; 8 waves x 64 cols; two 512x512 WMMA GEMMs.
// ---------------------------------------------------------------------------
__global__ void __launch_bounds__(256) build_temb_table(
    const float* __restrict__ te_w1, const float* __restrict__ te_b1,
    const float* __restrict__ te_w2, const float* __restrict__ te_b2,
    float* __restrict__ table) {
  __shared__ float sh[16 * AST];
  const int t = threadIdx.x;
  const int wid = t >> 5, lane = t & 31, tid = lane & 15, half = lane >> 4;
  const int r0 = blockIdx.x * 16;
  const float lf = -logf(10000.0f) / 256.0f;

  // sinusoidal features E[16][512]: [sin(t*f_i) | cos(t*f_i)], f_i = exp(-ln(1e4)*i/256)
  for (int i = t; i < 16 * 512; i += 256) {
    int m = i >> 9, d = i & 511;
    float tv = (float)(r0 + m);
    float v = (d < 256) ? sinf(tv * expf(lf * (float)d))
                        : cosf(tv * expf(lf * (float)(d - 256)));
    sh[m * AST + d] = v;
  }
  __syncthreads();

  const int colbase = wid * 64;
  const v8f vzero = {0.f, 0.f, 0.f, 0.f, 0.f, 0.f, 0.f, 0.f};
  v8f acc[4];
  for (int ct = 0; ct < 4; ++ct) acc[ct] = vzero;

  // H = gelu(E @ te_w1 + b1)
  for (int k = 0; k < 512; k += 4) {
    int kk = k + 2 * half;
    v2f a; a.x = sh[tid * AST + kk]; a.y = sh[tid * AST + kk + 1];
    for (int ct = 0; ct < 4; ++ct) {
      int col = colbase + ct * 16 + tid;
      v2f b; b.x = te_w1[kk * 512 + col]; b.y = te_w1[(kk + 1) * 512 + col];
      acc[ct] = wmma4(a, b, acc[ct]);
    }
  }
  __syncthreads();                      // everyone done reading E
  for (int ct = 0; ct < 4; ++ct) {
    int col = colbase + ct * 16 + tid;
    float bias = te_b1[col];
    for (int r = 0; r < 8; ++r) {
      int m = r + 8 * half;
      sh[m * AST + col] = gelu_exact(acc[ct][r] + bias);
    }
  }
  __syncthreads();

  // OUT = H @ te_w2 + b2
  for (int ct = 0; ct < 4; ++ct) acc[ct] = vzero;
  for (int k = 0; k < 512; k += 4) {
    int kk = k + 2 * half;
    v2f a; a.x = sh[tid * AST + kk]; a.y = sh[tid * AST + kk + 1];
    for (int ct = 0; ct < 4; ++ct) {
      int col = colbase + ct * 16 + tid;
      v2f b; b.x = te_w2[kk * 512 + col]; b.y = te_w2[(kk + 1) * 512 + col];
      acc[ct] = wmma4(a, b, acc[ct]);
    }
  }
  for (int ct = 0; ct < 4; ++ct) {
    int col = colbase + ct * 16 + tid;
    float bias = te_b2[col];
    for (int r = 0; r < 8; ++r) {
      int m = r + 8 * half;
      table[(size_t)(r0 + m) * 512 + col] = acc[ct][r] + bias;
    }
  }
}

// ---------------------------------------------------------------------------
// K1: ragged per-segment mean pairwise distance on pos[:,:,1,:].
// Segment = (cls .. eos) exclusive; walk left for cls, right for eos/cls end.
// ---------------------------------------------------------------------------
__global__ void __launch_bounds__(256) dist_kernel(
    const float* __restrict__ pos, const int* __restrict__ aa,
    float* __restrict__ dist) {
  int g = blockIdx.x * blockDim.x + threadIdx.x;
  if (g >= BS * L) return;
  int b = g / L, i = g % L;
  const int* a = aa + b * L;
  int tok = a[i];
  float out = 0.0f;
  if (tok != 0 && tok != 2) {
    int s = -1;
    for (int j = i - 1; j >= 0; --j) {       // left: cls before eos => in a segment
      int tj = a[j];
      if (tj == 0) { s = j; break; }
      if (tj == 2) break;
    }
    if (s >= 0) {
      int e = L;
      for (int j = i + 1; j < L; ++j) {      // right: segment ends at eos or next cls
        int tj = a[j];
        if (tj == 2 || tj == 0) { e = j; break; }
      }
      const float* p = pos + (size_t)(b * L) * 9 + 3;   // atom index 1
      float xi = p[(size_t)i * 9 + 0];
      float yi = p[(size_t)i * 9 + 1];
      float zi = p[(size_t)i * 9 + 2];
      float sum = 0.0f;
      for (int j = s + 1; j < e; ++j) {
        float dx = xi - p[(size_t)j * 9 + 0];
        float dy = yi - p[(size_t)j * 9 + 1];
        float dz = zi - p[(size_t)j * 9 + 2];
        float sq = dx * dx + dy * dy + dz * dz;
        sum += (sq > 0.0f) ? sqrtf(sq) : 0.0f;
      }
      int cnt = e - s - 1;
      out = sum / (float)(cnt < 1 ? 1 : cnt);
    }
  }
  dist[g] = out;
}

// ---------------------------------------------------------------------------
// K2: fused angle/pos feature build + projections. One block = 16 seq rows.
// out[b,l, 0:512]   = (angle_feat rows 6l..6l+5, 3072-K) @ angproj_w + b
// out[b,l, 512:1024]= pos_feat @ posproj_w + b ; then cls/eos/pad overrides.
// ---------------------------------------------------------------------------
__global__ void __launch_bounds__(256) fused_kernel(
    const float* __restrict__ angle,       // [BS,L,6]
    const int* __restrict__ padding_mask,  // [BS,L]
    const int* __restrict__ mask_angle,    // [BS,L,1]
    const int* __restrict__ angle_mask,    // [BS,L,3]
    const int* __restrict__ bond_mask,     // [BS,L,3]
    const float* __restrict__ time_pos,    // [BS,L]
    const int* __restrict__ aa,            // [BS,L]
    const float* __restrict__ ang_w1,      // [128]
    const float* __restrict__ ang_w2,      // [128,512]
    const float* __restrict__ pos_w1,      // [1]
    const float* __restrict__ pos_w2,      // [512]
    const float* __restrict__ angproj_w,   // [3072,512]
    const float* __restrict__ angproj_b,   // [512]
    const float* __restrict__ posproj_w,   // [512,512]
    const float* __restrict__ posproj_b,   // [512]
    const float* __restrict__ cls_w,       // [1024]
    const float* __restrict__ eos_w,       // [1024]
    const float* __restrict__ unkang,      // [512]
    const float* __restrict__ unkpos,      // [512]
    const float* __restrict__ table,       // [1024,512]
    const float* __restrict__ dist,        // [BS*L]
    float* __restrict__ out) {             // [BS,L,1024]
  __shared__ float shA[16 * AST];
  __shared__ float shH[16 * HST];
  __shared__ float sAng[16][6];
  __shared__ int   sAmask[16][6];
  __shared__ int   sIdxA[16][6];
  __shared__ int   sTok[16];
  __shared__ int   sPad[16];
  __shared__ int   sPosIdx[16];
  __shared__ float sPosG[16];

  const int t = threadIdx.x;
  const int wid = t >> 5, lane = t & 31, tid = lane & 15, half = lane >> 4;
  const int m0 = blockIdx.x * 16;
  const int b = m0 / L, l0 = m0 % L;
  const int base = b * L;

  if (t < 16) {
    int l = l0 + t;
    int tok = aa[base + l];
    sTok[t] = tok;
    sPad[t] = padding_mask[base + l];
    int tmask = (mask_angle[base + l] != 0) && (tok != 0) && (tok != 2);
    int tint = (int)(time_pos[base + l] + 0.5f);
    if (tint < 0) tint = 0;
    if (tint > TROWS - 1) tint = TROWS - 1;
    sPosIdx[t] = tmask ? tint : 0;
    sPosG[t] = gelu_exact(dist[base + l] * pos_w1[0]);
  }
  if (t < 96) {
    int m = t / 6, c = t % 6;
    int l = l0 + m;
    int am = (c < 3) ? angle_mask[(base + l) * 3 + c]
                     : bond_mask[(base + l) * 3 + (c - 3)];
    sAmask[m][c] = (am != 0);
    sAng[m][c] = (am != 0) ? angle[(base + l) * 6 + c] : 0.0f;
    // temb_ang select-mask uses torch .repeat(1,6,1): index (6l+c) % L
    int lm = (6 * l + c) % L;
    int tokm = aa[base + lm];
    int tmaskm = (mask_angle[base + lm] != 0) && (tokm != 0) && (tokm != 2);
    int tint = (int)(time_pos[base + l] + 0.5f);   // ta value comes from row l
    if (tint < 0) tint = 0;
    if (tint > TROWS - 1) tint = TROWS - 1;
    sIdxA[m][c] = (tmaskm && (am != 0)) ? tint : 0;   // ta==0 when !amask -> t0
  }
  __syncthreads();

  const int colbase = wid * 64;
  const v8f vzero = {0.f, 0.f, 0.f, 0.f, 0.f, 0.f, 0.f, 0.f};
  v8f accO[4];
  for (int ct = 0; ct < 4; ++ct) accO[ct] = vzero;

  // ---- angle half: K = 6 chunks x 512 ----
  for (int c = 0; c < 6; ++c) {
    // h[16][128] = gelu(ang * ang_w1) (0 if masked; replaced by unkang below)
    for (int i = t; i < 16 * 128; i += 256) {
      int m = i >> 7, j = i & 127;
      float hv = sAmask[m][c] ? gelu_exact(sAng[m][c] * ang_w1[j]) : 0.0f;
      shH[m * HST + j] = hv;
    }
    __syncthreads();
    // Asub[16][512] = (amask ? h @ ang_w2 : unkang) + table[idx]
    for (int ct = 0; ct < 4; ++ct) {
      int col = colbase + ct * 16 + tid;
      v8f mv = vzero;
      for (int k = 0; k < 128; k += 4) {
        int kk = k + 2 * half;
        v2f a; a.x = shH[tid * HST + kk]; a.y = shH[tid * HST + kk + 1];
        v2f bb; bb.x = ang_w2[kk * 512 + col]; bb.y = ang_w2[(kk + 1) * 512 + col];
        mv = wmma4(a, bb, mv);
      }
      float uv = unkang[col];
      for (int r = 0; r < 8; ++r) {
        int m = r + 8 * half;
        float v = sAmask[m][c] ? mv[r] : uv;
        v += table[(size_t)sIdxA[m][c] * 512 + col];
        shA[m * AST + col] = v;
      }
    }
    __syncthreads();
    // accumulate this K-chunk of the angproj GEMM
    const float* Bp = angproj_w + (size_t)c * 512 * 512;
    for (int k = 0; k < 512; k += 4) {
      int kk = k + 2 * half;
      v2f a; a.x = shA[tid * AST + kk]; a.y = shA[tid * AST + kk + 1];
      for (int ct = 0; ct < 4; ++ct) {
        int col = colbase + ct * 16 + tid;
        v2f bb; bb.x = Bp[kk * 512 + col]; bb.y = Bp[(kk + 1) * 512 + col];
        accO[ct] = wmma4(a, bb, accO[ct]);
      }
    }
    __syncthreads();
  }
  // store angle half (cols 0..511) with overrides
  for (int ct = 0; ct < 4; ++ct) {
    int col = colbase + ct * 16 + tid;
    float bias = angproj_b[col];
    for (int r = 0; r < 8; ++r) {
      int m = r + 8 * half;
      float v = accO[ct][r] + bias;
      int tok = sTok[m];
      if (tok == 0) v = cls_w[col];
      else if (tok == 2) v = eos_w[col];
      if (sPad[m]) v = 0.0f;
      out[(size_t)(base + l0 + m) * ED + col] = v;
    }
  }

  // ---- pos half ----
  for (int i = t; i < 16 * 512; i += 256) {
    int m = i >> 9, d = i & 511;
    int tok = sTok[m];
    float v = (tok == 0 || tok == 2) ? unkpos[d] : sPosG[m] * pos_w2[d];
    v += table[(size_t)sPosIdx[m] * 512 + d];
    shA[m * AST + d] = v;
  }
  __syncthreads();
  v8f accP[4];
  for (int ct = 0; ct < 4; ++ct) accP[ct] = vzero;
  for (int k = 0; k < 512; k += 4) {
    int kk = k + 2 * half;
    v2f a; a.x = shA[tid * AST + kk]; a.y = shA[tid * AST + kk + 1];
    for (int ct = 0; ct < 4; ++ct) {
      int col = colbase + ct * 16 + tid;
      v2f bb; bb.x = posproj_w[kk * 512 + col]; bb.y = posproj_w[(kk + 1) * 512 + col];
      accP[ct] = wmma4(a, bb, accP[ct]);
    }
  }
  for (int ct = 0; ct < 4; ++ct) {
    int col = colbase + ct * 16 + tid;
    float bias = posproj_b[col];
    for (int r = 0; r < 8; ++r) {
      int m = r + 8 * half;
      float v = accP[ct][r] + bias;
      int tok = sTok[m];
      if (tok == 0) v = cls_w[512 + col];
      else if (tok == 2) v = eos_w[512 + col];
      if (sPad[m]) v = 0.0f;
      out[(size_t)(base + l0 + m) * ED + 512 + col] = v;
    }
  }
}

// ---------------------------------------------------------------------------
extern "C" void kernel_launch(void* const* d_in, const int* in_sizes, int n_in,
                              void* d_out, int out_size, void* d_ws, size_t ws_size,
                              hipStream_t stream) {
  const float* pos        = (const float*)d_in[0];
  const float* angle      = (const float*)d_in[1];
  const int*   padding    = (const int*)d_in[2];
  // d_in[3] mask_aa, d_in[4] mask_pos: unused by the reference output
  const int*   mask_ang   = (const int*)d_in[5];
  const int*   angle_mask = (const int*)d_in[6];
  const int*   bond_mask  = (const int*)d_in[7];
  const float* time_pos   = (const float*)d_in[8];
  // d_in[9] time_angle: overwritten in reference, unused
  const int*   aa         = (const int*)d_in[10];
  const float* ang_w1     = (const float*)d_in[11];
  const float* ang_w2     = (const float*)d_in[12];
  const float* pos_w1     = (const float*)d_in[13];
  const float* pos_w2     = (const float*)d_in[14];
  const float* te_w1      = (const float*)d_in[15];
  const float* te_b1      = (const float*)d_in[16];
  const float* te_w2      = (const float*)d_in[17];
  const float* te_b2      = (const float*)d_in[18];
  const float* angproj_w  = (const float*)d_in[19];
  const float* angproj_b  = (const float*)d_in[20];
  const float* posproj_w  = (const float*)d_in[21];
  const float* posproj_b  = (const float*)d_in[22];
  const float* cls_w      = (const float*)d_in[23];
  const float* eos_w      = (const float*)d_in[24];
  const float* unkang     = (const float*)d_in[25];
  const float* unkpos     = (const float*)d_in[26];

  float* table = (float*)d_ws;               // 1024*512 f32 = 2 MB
  float* dst   = table + TROWS * D;          // BS*L f32     = 32 KB
  float* outp  = (float*)d_out;

  hipLaunchKernelGGL(build_temb_table, dim3(TROWS / 16), dim3(256), 0, stream,
                     te_w1, te_b1, te_w2, te_b2, table);
  hipLaunchKernelGGL(dist_kernel, dim3((BS * L + 255) / 256), dim3(256), 0, stream,
                     pos, aa, dst);
  hipLaunchKernelGGL(fused_kernel, dim3(BS * L / 16), dim3(256), 0, stream,
                     angle, padding, mask_ang, angle_mask, bond_mask, time_pos,
                     aa, ang_w1, ang_w2, pos_w1, pos_w2, angproj_w, angproj_b,
                     posproj_w, posproj_b, cls_w, eos_w, unkang, unkpos,
                     table, dst, outp);
}